// FiLMAtten_4217657884987
// MI455X (gfx1250) — compile-verified
//
#include <hip/hip_runtime.h>
#include <hip/hip_bf16.h>

// FiLM channel-attention for MI455X (gfx1250, wave32).
// fp32 end-to-end (problem is HBM-bound: ~0.7GB traffic @23.3TB/s >> 30 GFLOP
// of fp32 WMMA work), all matmuls via V_WMMA_F32_16X16X4_F32.
// All LDS operands are stored so each lane's K-pair is a contiguous,
// 8B-aligned b64 load (no VGPR repacking in front of the WMMAs).

#define BB 16
#define CC 256
#define HH 64
#define WW 64
#define NN 4096          // H*W
#define HEADS 8
#define HS 32            // head size
#define THREEC 768
#define OUTC 512         // concat output channels
#define CHUNK 256        // attention n-chunk
#define NCHUNK (NN / CHUNK)

typedef float v2f __attribute__((ext_vector_type(2)));
typedef float v8f __attribute__((ext_vector_type(8)));

__device__ __forceinline__ float silu_f(float v) {
    return v * (1.0f / (1.0f + __expf(-v)));
}

__device__ __forceinline__ v8f wmma_f32_4(v2f a, v2f b, v8f c) {
    return __builtin_amdgcn_wmma_f32_16x16x4_f32(
        /*neg_a=*/false, a, /*neg_b=*/false, b,
        /*c_mod=*/(short)0, c, /*reuse_a=*/false, /*reuse_b=*/false);
}

// ---------------------------------------------------------------------------
// Kernel 1: x_preact = silu(x + pre_act_bias) -> out[:, 0:C, :, :]
// ---------------------------------------------------------------------------
__global__ __launch_bounds__(256) void preact_kernel(
    const float* __restrict__ x, const float* __restrict__ pab,
    float* __restrict__ out) {
    long i4 = (long)blockIdx.x * blockDim.x + threadIdx.x; // over B*C*N/4
    long i = i4 * 4;
    int n = (int)(i % NN);
    int c = (int)((i / NN) % CC);
    int b = (int)(i / ((long)NN * CC));
    float4 xv = ((const float4*)x)[i4];
    float bias = pab[c];
    float4 r;
    r.x = silu_f(xv.x + bias);
    r.y = silu_f(xv.y + bias);
    r.z = silu_f(xv.z + bias);
    r.w = silu_f(xv.w + bias);
    *(float4*)&out[(((long)b * OUTC + c) * NN + n)] = r;
}

// ---------------------------------------------------------------------------
// Kernel 2: qkv[b] = qkv_w[768,256] @ (x_preact[b] + pre_qkv_bias) + qkv_b
// 128x128 tile per WG, 8 waves, f32 WMMA 16x16x4, K staged through LDS.
// A tile stored [m][k], B tile stored K-major [n][k]: every fragment is a
// contiguous aligned b64 LDS load.
// ---------------------------------------------------------------------------
#define BM 128
#define BN 128
#define BK 32
#define KPAD 34   // row stride (floats): 34*l mod 64 banks distinct per half

__global__ __launch_bounds__(256) void qkv_gemm_kernel(
    const float* __restrict__ outbuf,   // x_preact lives in out[:, 0:C]
    const float* __restrict__ qkv_w, const float* __restrict__ qkv_b,
    const float* __restrict__ pqb, float* __restrict__ qkv) {
    __shared__ float Wl[BM][KPAD];      // [m][k]
    __shared__ float Xl[BN][KPAD];      // [n][k]  (K-major!)

    const int b  = blockIdx.z;
    const int m0 = blockIdx.x * BM;     // 6 tiles over 768
    const int n0 = blockIdx.y * BN;     // 32 tiles over 4096
    const int tid  = threadIdx.x;
    const int lane = tid & 31;
    const int wave = tid >> 5;
    const int lrow = lane & 15;
    const int ksel = (lane >> 4) * 2;   // lane-half selects K pair

    const float* xp = outbuf + (long)b * OUTC * NN; // channels 0..255

    v8f acc[8];
    for (int t = 0; t < 8; ++t)
        for (int e = 0; e < 8; ++e) acc[t][e] = 0.0f;

    for (int k0 = 0; k0 < CC; k0 += BK) {
        __syncthreads();
        // stage W tile: 128x32, float4 along K (contiguous in global)
        for (int s = tid; s < BM * BK / 4; s += 256) {
            int row = s / (BK / 4);
            int kc  = (s % (BK / 4)) * 4;
            float4 wv = *(const float4*)&qkv_w[(long)(m0 + row) * CC + k0 + kc];
            Wl[row][kc + 0] = wv.x; Wl[row][kc + 1] = wv.y;
            Wl[row][kc + 2] = wv.z; Wl[row][kc + 3] = wv.w;
        }
        // stage X tile: global [k][n] float4 along n, transpose into [n][k]
        for (int s = tid; s < BK * BN / 4; s += 256) {
            int kk = s / (BN / 4);
            int nc = (s % (BN / 4)) * 4;
            float4 xv = *(const float4*)&xp[(long)(k0 + kk) * NN + n0 + nc];
            float pb = pqb[k0 + kk];
            Xl[nc + 0][kk] = xv.x + pb;
            Xl[nc + 1][kk] = xv.y + pb;
            Xl[nc + 2][kk] = xv.z + pb;
            Xl[nc + 3][kk] = xv.w + pb;
        }
        __syncthreads();

        const int mrow = wave * 16 + lrow;
        for (int kk = 0; kk < BK; kk += 4) {
            v2f a = *(const v2f*)&Wl[mrow][kk + ksel];
            for (int nt = 0; nt < 8; ++nt) {
                int col = nt * 16 + lrow;
                v2f bf = *(const v2f*)&Xl[col][kk + ksel];
                acc[nt] = wmma_f32_4(a, bf, acc[nt]);
            }
        }
    }

    // epilogue: D layout => VGPR r holds M = r (lanes<16) / r+8 (lanes>=16)
    const int mbase = m0 + wave * 16 + (lane >> 4) * 8;
    for (int nt = 0; nt < 8; ++nt) {
        int col = n0 + nt * 16 + lrow;
        for (int r = 0; r < 8; ++r) {
            int m = mbase + r;
            qkv[((long)b * THREEC + m) * NN + col] = acc[nt][r] + qkv_b[m];
        }
    }
}

// ---------------------------------------------------------------------------
// Kernel 3: partial logits. Per (b,h,chunk): load q,k chunk with PE + FiLM
// fused, compute 32x32 Gram partial via 4 waves of 16x16x4 WMMA.
// Both operands are K(n)-major rows, so fragments are aligned b64 loads.
// ---------------------------------------------------------------------------
#define QPAD (CHUNK + 2)   // even pad: keeps lane K-pairs 8B-aligned

__global__ __launch_bounds__(128) void logits_kernel(
    const float* __restrict__ qkv,
    const float* __restrict__ mm, const float* __restrict__ mb,
    const float* __restrict__ peqh, const float* __restrict__ peqw,
    const float* __restrict__ pekh, const float* __restrict__ pekw,
    float* __restrict__ partial) {
    __shared__ float ql[HS][QPAD];
    __shared__ float kl[HS][QPAD];

    const int bh = blockIdx.y;
    const int b  = bh >> 3;
    const int h  = bh & 7;
    const int n0 = blockIdx.x * CHUNK;
    const int tid  = threadIdx.x;
    const int lane = tid & 31;
    const int wave = tid >> 5;

    // stage q,k (32 x 256) with positional encodings + modulation fused
    for (int s = tid; s < HS * CHUNK / 4; s += 128) {
        int c  = s / (CHUNK / 4);
        int nc = (s % (CHUNK / 4)) * 4;
        int n  = n0 + nc;
        int y  = n >> 6;           // n..n+3 share y
        int xw = n & 63;
        int cg = h * HS + c;       // channel index within C
        float4 qv = *(const float4*)&qkv[((long)b * THREEC + cg) * NN + n];
        float4 kv = *(const float4*)&qkv[((long)b * THREEC + CC + cg) * NN + n];
        float qh = peqh[cg * HH + y];
        float kh = pekh[cg * HH + y];
        float mu = mm[b * CC + cg];
        float be = mb[b * CC + cg];
        ql[c][nc + 0] = (qv.x + qh + peqw[cg * WW + xw + 0]) * mu + be;
        ql[c][nc + 1] = (qv.y + qh + peqw[cg * WW + xw + 1]) * mu + be;
        ql[c][nc + 2] = (qv.z + qh + peqw[cg * WW + xw + 2]) * mu + be;
        ql[c][nc + 3] = (qv.w + qh + peqw[cg * WW + xw + 3]) * mu + be;
        kl[c][nc + 0] = kv.x + kh + pekw[cg * WW + xw + 0];
        kl[c][nc + 1] = kv.y + kh + pekw[cg * WW + xw + 1];
        kl[c][nc + 2] = kv.z + kh + pekw[cg * WW + xw + 2];
        kl[c][nc + 3] = kv.w + kh + pekw[cg * WW + xw + 3];
    }
    __syncthreads();

    // 4 waves -> 4 tiles of the 32x32 logits partial
    const int ti = wave >> 1;
    const int tj = wave & 1;
    const int lrow = lane & 15;
    const int ksel = (lane >> 4) * 2;

    v8f acc;
    for (int e = 0; e < 8; ++e) acc[e] = 0.0f;

    for (int kk = 0; kk < CHUNK; kk += 4) {
        v2f a  = *(const v2f*)&ql[ti * 16 + lrow][kk + ksel];
        v2f bf = *(const v2f*)&kl[tj * 16 + lrow][kk + ksel];
        acc = wmma_f32_4(a, bf, acc);
    }

    float* p = partial + ((long)bh * NCHUNK + blockIdx.x) * (HS * HS);
    const int row0 = ti * 16 + (lane >> 4) * 8;
    for (int r = 0; r < 8; ++r)
        p[(row0 + r) * HS + tj * 16 + lrow] = acc[r];
}

// ---------------------------------------------------------------------------
// Kernel 4: reduce partials, scale by 1/sqrt(hs), row softmax (wave32/row).
// ---------------------------------------------------------------------------
__global__ __launch_bounds__(256) void softmax_kernel(
    const float* __restrict__ partial, float* __restrict__ weights) {
    const int lane = threadIdx.x & 31;
    const int wave = threadIdx.x >> 5;
    const int row  = blockIdx.x * 8 + wave;   // B*HEADS*HS = 4096 rows
    const int bh = row >> 5;
    const int c  = row & 31;

    float s = 0.0f;
    for (int ch = 0; ch < NCHUNK; ++ch)
        s += partial[((long)bh * NCHUNK + ch) * (HS * HS) + c * HS + lane];
    s *= 0.17677669529663687f;  // 1/sqrt(32)

    float mx = s;
    for (int o = 16; o > 0; o >>= 1) mx = fmaxf(mx, __shfl_xor(mx, o, 32));
    float e = __expf(s - mx);
    float sum = e;
    for (int o = 16; o > 0; o >>= 1) sum += __shfl_xor(sum, o, 32);
    weights[(long)bh * (HS * HS) + c * HS + lane] = e / sum;
}

// ---------------------------------------------------------------------------
// Kernel 5: out = weights[32,32] @ v[32,chunk] -> out[:, C:2C, :, :]
// v staged K-major [n][k] so B fragments are aligned b64 loads.
// ---------------------------------------------------------------------------
__global__ __launch_bounds__(256) void out_gemm_kernel(
    const float* __restrict__ qkv, const float* __restrict__ weights,
    float* __restrict__ out) {
    __shared__ float wl[HS][KPAD];      // [m][k]
    __shared__ float vl[CHUNK][KPAD];   // [n][k]  (K-major!)

    const int bh = blockIdx.y;
    const int b  = bh >> 3;
    const int h  = bh & 7;
    const int n0 = blockIdx.x * CHUNK;
    const int tid  = threadIdx.x;
    const int lane = tid & 31;
    const int wave = tid >> 5;

    for (int s = tid; s < HS * HS; s += 256)
        wl[s >> 5][s & 31] = weights[(long)bh * (HS * HS) + s];
    for (int s = tid; s < HS * CHUNK / 4; s += 256) {
        int c  = s / (CHUNK / 4);
        int nc = (s % (CHUNK / 4)) * 4;
        float4 vv = *(const float4*)
            &qkv[((long)b * THREEC + 2 * CC + h * HS + c) * NN + n0 + nc];
        vl[nc + 0][c] = vv.x;
        vl[nc + 1][c] = vv.y;
        vl[nc + 2][c] = vv.z;
        vl[nc + 3][c] = vv.w;
    }
    __syncthreads();

    // 8 waves: ti = M-tile (0..1), 4 N-tiles each
    const int ti  = wave >> 2;
    const int tjb = (wave & 3) * 4;
    const int lrow = lane & 15;
    const int ksel = (lane >> 4) * 2;

    v8f acc[4];
    for (int t = 0; t < 4; ++t)
        for (int e = 0; e < 8; ++e) acc[t][e] = 0.0f;

    for (int kk = 0; kk < HS; kk += 4) {
        v2f a = *(const v2f*)&wl[ti * 16 + lrow][kk + ksel];
        for (int t = 0; t < 4; ++t) {
            int col = (tjb + t) * 16 + lrow;
            v2f bf = *(const v2f*)&vl[col][kk + ksel];
            acc[t] = wmma_f32_4(a, bf, acc[t]);
        }
    }

    const int row0 = ti * 16 + (lane >> 4) * 8;
    for (int t = 0; t < 4; ++t) {
        int col = n0 + (tjb + t) * 16 + lrow;
        for (int r = 0; r < 8; ++r) {
            int m = row0 + r;
            out[((long)b * OUTC + CC + h * HS + m) * NN + col] = acc[t][r];
        }
    }
}

// ---------------------------------------------------------------------------
extern "C" void kernel_launch(void* const* d_in, const int* in_sizes, int n_in,
                              void* d_out, int out_size, void* d_ws, size_t ws_size,
                              hipStream_t stream) {
    const float* x    = (const float*)d_in[0];
    const float* mm   = (const float*)d_in[1];
    const float* mb   = (const float*)d_in[2];
    const float* qkvw = (const float*)d_in[3];
    const float* qkvb = (const float*)d_in[4];
    const float* peqh = (const float*)d_in[5];
    const float* peqw = (const float*)d_in[6];
    const float* pekh = (const float*)d_in[7];
    const float* pekw = (const float*)d_in[8];
    const float* pab  = (const float*)d_in[9];
    const float* pqb  = (const float*)d_in[10];
    float* out = (float*)d_out;

    float* ws      = (float*)d_ws;
    float* qkv     = ws;                                        // 16*768*4096
    float* partial = qkv + (long)BB * THREEC * NN;              // 128*16*1024
    float* wts     = partial + (long)BB * HEADS * NCHUNK * HS * HS; // 128*1024

    // 1) silu preact -> out[:, 0:C]
    preact_kernel<<<(BB * CC * NN / 4) / 256, 256, 0, stream>>>(x, pab, out);
    // 2) QKV GEMM (reads x_preact from out, fuses pre_qkv_bias & qkv_b)
    qkv_gemm_kernel<<<dim3(THREEC / BM, NN / BN, BB), 256, 0, stream>>>(
        out, qkvw, qkvb, pqb, qkv);
    // 3) partial logits with PE + FiLM fused
    logits_kernel<<<dim3(NCHUNK, BB * HEADS), 128, 0, stream>>>(
        qkv, mm, mb, peqh, peqw, pekh, pekw, partial);
    // 4) softmax
    softmax_kernel<<<(BB * HEADS * HS) / 8, 256, 0, stream>>>(partial, wts);
    // 5) weights @ v -> out[:, C:2C]
    out_gemm_kernel<<<dim3(NCHUNK, BB * HEADS), 256, 0, stream>>>(qkv, wts, out);
}